// loss_generator_58420145160558
// MI455X (gfx1250) — compile-verified
//
#include <hip/hip_runtime.h>
#include <hip/hip_bf16.h>

// FDTD wave-equation rollout for MI455X (gfx1250).
// One kernel launch per time step (strict sequential dependency between steps).
// Each block computes a 64x16 output tile; wave 0 stages the 66x18 halo tile
// of `cur` into LDS with a single Tensor Data Mover DMA (tensor_load_to_lds),
// waits on TENSORcnt, barriers, then all 8 waves compute the stencil +
// absorbing boundaries from LDS. 4 rows per thread amortize per-block SALU.

typedef __attribute__((ext_vector_type(4))) unsigned int v4u;
typedef __attribute__((ext_vector_type(8))) int          v8i;
typedef __attribute__((ext_vector_type(4))) int          v4i;

#define TILE_W 64
#define TILE_H 16
#define HALO_W 66   // TILE_W + 2
#define HALO_H 18   // TILE_H + 2

__global__ __launch_bounds__(256)
void fdtd_step_kernel(const float* __restrict__ prev,
                      const float* __restrict__ cur,
                      const float* __restrict__ cmap,
                      const int*   __restrict__ locx,
                      const int*   __restrict__ locy,
                      const int*   __restrict__ pbs1,
                      const int*   __restrict__ pid2,
                      float*       __restrict__ outf,
                      int n, int W, int H, int n_src)
{
    __shared__ float sh[HALO_H * HALO_W];

    const int tx = threadIdx.x;              // 0..63  (column within tile)
    const int ty = threadIdx.y;              // 0..3   (row group)
    const int gj  = blockIdx.x * TILE_W + tx;          // global column
    const int gi0 = blockIdx.y * TILE_H + ty * 4;      // first of 4 rows

    // Halo-tile origin, clamped fully in-bounds.
    int org_r = blockIdx.y * TILE_H - 1;
    if (org_r < 0) org_r = 0;
    if (org_r > H - HALO_H) org_r = H - HALO_H;
    int org_c = blockIdx.x * TILE_W - 1;
    if (org_c < 0) org_c = 0;
    if (org_c > W - HALO_W) org_c = W - HALO_W;

    // Wave 0 issues a single TDM DMA: 66x18 f32 tile of `cur` -> LDS.
    const int lin = ty * TILE_W + tx;
    if (lin < 32) {
        unsigned lds_off = (unsigned)(unsigned long long)(&sh[0]);
        const float* gptr = cur + ((long long)org_r * W + org_c);
        unsigned long long gaddr = (unsigned long long)gptr;

        // D# group 0: count=1 | lds_addr | global_addr[56:0] | type=2
        v4u g0 = { 1u,
                   lds_off,
                   (unsigned)(gaddr & 0xFFFFFFFFull),
                   (unsigned)((gaddr >> 32) & 0x01FFFFFFull) | (2u << 30) };

        // D# group 1: data_size=2 (4B), tensor_dim0=W, tensor_dim1=H,
        // tile_dim0=66, tile_dim1=18, tensor_dim0_stride=W (elements).
        const unsigned td0 = (unsigned)W, td1 = (unsigned)H;
        const unsigned t0 = HALO_W, t1 = HALO_H;
        const unsigned long long s0 = (unsigned long long)W;
        v8i g1 = { (int)(2u << 16),                                  // data_size=4B
                   (int)((td0 & 0xFFFFu) << 16),                     // td0[15:0]
                   (int)((td0 >> 16) | ((td1 & 0xFFFFu) << 16)),     // td0[31:16]|td1[15:0]
                   (int)((td1 >> 16) | (t0 << 16)),                  // td1[31:16]|tile0
                   (int)(t1),                                        // tile1 | tile2=0
                   (int)(unsigned)(s0 & 0xFFFFFFFFull),              // stride0[31:0]
                   (int)(unsigned)(s0 >> 32),                        // stride0[47:32]
                   0 };
        v4i g2 = { 0, 0, 0, 0 };              // 2-D tensor: groups 2/3 unused
        v4i g3 = { 0, 0, 0, 0 };
        v8i g4 = { 0, 0, 0, 0, 0, 0, 0, 0 };  // extra group (clang-23 6-arg form)

        __builtin_amdgcn_tensor_load_to_lds(g0, g1, g2, g3, g4, 0);
        __builtin_amdgcn_s_wait_tensorcnt(0);
    }
    __syncthreads();

    const float dt = 1e-4f;
    const float dx = 1.0f;
    const int   lj = gj - org_c;

    #pragma unroll
    for (int m = 0; m < 4; ++m) {
        const int gi = gi0 + m;
        const int li = gi - org_r;
        const float cc  = cmap[(long long)gi * W + gj];
        const float cen = sh[li * HALO_W + lj];

        float val;
        // Priority = reverse of reference's sequential overwrites:
        // row H-1 (last applied) > col W-1 > col 0 > row 0 > source > interior.
        if (gi == H - 1) {
            float c1 = sh[(li - 1) * HALO_W + lj];
            val = cen - dt * cc * (cen - c1) / dx;
        } else if (gj == W - 1) {
            float c1 = sh[li * HALO_W + (lj - 1)];
            val = cen - dt * cc * (cen - c1) / dx;
        } else if (gj == 0) {
            float c1 = sh[li * HALO_W + (lj + 1)];
            val = cen - dt * cc * (cen - c1) / dx;
        } else if (gi == 0) {
            float c1 = sh[(li + 1) * HALO_W + lj];
            val = cen - dt * cc * (cen - c1) / dx;
        } else {
            const float pv = prev[(long long)gi * W + gj];
            const float up = sh[(li - 1) * HALO_W + lj];
            const float dn = sh[(li + 1) * HALO_W + lj];
            const float lf = sh[li * HALO_W + (lj - 1)];
            const float rt = sh[li * HALO_W + (lj + 1)];
            const float r = (dt * dt) * (cc * cc) / (dx * dx);
            const float lap = r * ((dn - 2.0f * cen + up) + (rt - 2.0f * cen + lf));
            val = 2.0f * cen - pv + lap;
            // Source injection (last matching source wins, like .at[].set).
            for (int k = 0; k < n_src; ++k) {
                int lx = locx[k];
                if (lx != -1 && lx == gi && locy[k] == gj) {
                    float t = (float)((*pbs1) * (*pid2) + n + 1);
                    val = 1500.0f * sinf(2.0f * 3.1415926f * 100.0f * t * dt);
                }
            }
        }
        outf[(long long)gi * W + gj] = val;
    }
}

extern "C" void kernel_launch(void* const* d_in, const int* in_sizes, int n_in,
                              void* d_out, int out_size, void* d_ws, size_t ws_size,
                              hipStream_t stream)
{
    const float* out0 = (const float*)d_in[0];  // [steps+2, 1, H, W] initial frames
    const float* cmap = (const float*)d_in[1];  // [1,1,H,W]
    const int*   locx = (const int*)d_in[2];    // [3]
    const int*   locy = (const int*)d_in[3];    // [3]
    const int*   bs1  = (const int*)d_in[4];    // scalar
    const int*   id2  = (const int*)d_in[5];    // scalar
    // d_in[6] = coffe (unused: steps derived from out_size)

    const int HW = in_sizes[1];
    int W = 1;
    while (W * W < HW) ++W;           // square grid (512)
    const int H = HW / W;
    const int steps = out_size / HW;  // coffe * bsize1
    const int n_src = in_sizes[2];

    float* out = (float*)d_out;
    dim3 block(TILE_W, 4);                       // 256 threads = 8 wave32
    dim3 grid(W / TILE_W, H / TILE_H);           // 8 x 32 = 256 blocks (512x512)

    for (int n = 1; n <= steps; ++n) {
        const float* prev = (n == 1) ? out0
                          : (n == 2) ? out0 + HW
                          : out + (size_t)(n - 3) * HW;
        const float* cur  = (n == 1) ? out0 + HW
                          : out + (size_t)(n - 2) * HW;
        float* newf = out + (size_t)(n - 1) * HW;
        fdtd_step_kernel<<<grid, block, 0, stream>>>(
            prev, cur, cmap, locx, locy, bs1, id2, newf, n, W, H, n_src);
    }
}